// _transpose_63513976373468
// MI455X (gfx1250) — compile-verified
//
#include <hip/hip_runtime.h>
#include <stdint.h>

// Static segment layout from the reference (_SEGMENTS):
//   (off, u, v): (0,128,1) (128,128,3) (512,128,5) (1152,128,7) (2048,64,9) (2624,64,11)
// Row length = 3328 f32. Output col p maps to input col off + (local%u)*v + (local/u),
// u is a power of two (128 or 64) -> shifts/masks only.

#define ROW_LEN    3328
#define ROW_VEC4   (ROW_LEN / 4)     // 832 = 3*256 + 64
#define BLK        256
#define NQ         4                 // vec4 slots per thread (4th valid only for tid<64)
#define ROWS_PER_BLOCK 8

__device__ __forceinline__ int perm_col(int p) {
    if (p < 128)  return p;                                                  // u=128, v=1 -> identity
    if (p < 512)  { int l = p - 128;  return 128  + (l & 127) * 3  + (l >> 7); }
    if (p < 1152) { int l = p - 512;  return 512  + (l & 127) * 5  + (l >> 7); }
    if (p < 2048) { int l = p - 1152; return 1152 + (l & 127) * 7  + (l >> 7); }
    if (p < 2624) { int l = p - 2048; return 2048 + (l & 63)  * 9  + (l >> 6); }
    {               int l = p - 2624; return 2624 + (l & 63)  * 11 + (l >> 6); }
}

// Issue this thread's slice of one row as async global->LDS b128 copies.
__device__ __forceinline__ void fill_async(const float* __restrict__ src,
                                           const float* lbuf, int tid) {
    const uint32_t lds_base = (uint32_t)(uintptr_t)lbuf;  // flat low 32 bits == LDS byte addr
#pragma unroll
    for (int j = 0; j < NQ; ++j) {
        const int q = tid + BLK * j;
        if (q < ROW_VEC4) {
            const uint32_t laddr = lds_base + 16u * (uint32_t)q;
            const void*    gaddr = (const void*)(src + 4 * q);
            asm volatile("global_load_async_to_lds_b128 %0, %1, off"
                         :
                         : "v"(laddr), "v"(gaddr)
                         : "memory");
        }
    }
}

// Wait for own async copies to land in LDS, then sync all waves in the workgroup.
__device__ __forceinline__ void async_barrier() {
    asm volatile("s_wait_asynccnt 0" ::: "memory");
    __syncthreads();
}

// Gather permuted columns from LDS with precomputed indices; coalesced b128 store.
__device__ __forceinline__ void drain(const float* lbuf, float* __restrict__ dst,
                                      const int idx[NQ][4], int tid) {
#pragma unroll
    for (int j = 0; j < NQ; ++j) {
        const int q = tid + BLK * j;
        if (q < ROW_VEC4) {
            float4 o;
            o.x = lbuf[idx[j][0]];
            o.y = lbuf[idx[j][1]];
            o.z = lbuf[idx[j][2]];
            o.w = lbuf[idx[j][3]];
            reinterpret_cast<float4*>(dst)[q] = o;
        }
    }
}

__global__ __launch_bounds__(BLK) void permute_rows_kernel(
    const float* __restrict__ x, float* __restrict__ out, int nrows)
{
    __shared__ float buf0[ROW_LEN];   // 13312 B
    __shared__ float buf1[ROW_LEN];   // 13312 B  (double buffer: 26.6 KB total)

    const int  tid  = threadIdx.x;
    const long base = (long)blockIdx.x * ROWS_PER_BLOCK;

    // Precompute this thread's gather indices ONCE; reused for all rows.
    int idx[NQ][4];
#pragma unroll
    for (int j = 0; j < NQ; ++j) {
        const int q = tid + BLK * j;
#pragma unroll
        for (int e = 0; e < 4; ++e)
            idx[j][e] = (q < ROW_VEC4) ? perm_col(4 * q + e) : 0;
    }

    const float* srcB = x   + base * ROW_LEN;
    float*       dstB = out + base * ROW_LEN;

    // Prologue: fill buf0 with row 0.
    if (base < nrows) fill_async(srcB, buf0, tid);
    async_barrier();

    // Software-pipelined, double-buffered row loop (buffer parity static).
#pragma unroll
    for (int k = 0; k < ROWS_PER_BLOCK; k += 2) {
        // buf0 holds row k; prefetch row k+1 into buf1 while draining.
        if (k + 1 < ROWS_PER_BLOCK && base + k + 1 < nrows)
            fill_async(srcB + (size_t)(k + 1) * ROW_LEN, buf1, tid);
        if (base + k < nrows)
            drain(buf0, dstB + (size_t)k * ROW_LEN, idx, tid);
        async_barrier();

        // buf1 holds row k+1; prefetch row k+2 into buf0 while draining.
        if (k + 2 < ROWS_PER_BLOCK && base + k + 2 < nrows)
            fill_async(srcB + (size_t)(k + 2) * ROW_LEN, buf0, tid);
        if (base + k + 1 < nrows)
            drain(buf1, dstB + (size_t)(k + 1) * ROW_LEN, idx, tid);
        async_barrier();
    }
}

extern "C" void kernel_launch(void* const* d_in, const int* in_sizes, int n_in,
                              void* d_out, int out_size, void* d_ws, size_t ws_size,
                              hipStream_t stream)
{
    const float* x = (const float*)d_in[0];
    float* out = (float*)d_out;

    const int rows    = in_sizes[0] / ROW_LEN;                       // 16384
    const int nblocks = (rows + ROWS_PER_BLOCK - 1) / ROWS_PER_BLOCK; // 2048
    permute_rows_kernel<<<dim3(nblocks), dim3(BLK), 0, stream>>>(x, out, rows);
}